// HSMNet_84885733638399
// MI455X (gfx1250) — compile-verified
//
#include <hip/hip_runtime.h>
#include <math.h>

// ---------------------------------------------------------------------------
// HSMNet disparity feature-volume, MI455X (gfx1250).
// out[b,c,d,h,x] = (x>=d) ? |ref[b,c,h,x] - tgt[b,c,h,x-d]| : 0
// Store-bandwidth bound (~215 MB out @ 23.3 TB/s ~ 9.2 us floor); inputs
// (10.5 MB) are staged row-wise into LDS once via CDNA5 async-tensor path
// (global_load_async_to_lds_b128 / ASYNCcnt) and reused up to D=48 times.
// Outputs leave via 128-bit non-temporal stores (215 MB > 192 MB L2, never
// re-read). Kernel templated on level constants so all index math is
// compile-time mul/shift (no runtime division).
// ---------------------------------------------------------------------------

typedef float v4f __attribute__((ext_vector_type(4)));
typedef __attribute__((address_space(3))) float lds_f;

__device__ __forceinline__ void async_copy16(const float* g, float* lds_generic)
{
  unsigned lds_addr = (unsigned)(unsigned long long)(lds_f*)lds_generic;
  asm volatile("global_load_async_to_lds_b128 %0, %1, off"
               :
               : "v"(lds_addr), "v"(g)
               : "memory");
}

__device__ __forceinline__ void wait_async0()
{
#if __has_builtin(__builtin_amdgcn_s_wait_asynccnt)
  __builtin_amdgcn_s_wait_asynccnt(0);
#else
  asm volatile("s_wait_asynccnt 0x0" ::: "memory");
#endif
}

template <int H, int W, int D, int ROWS>
__global__ __launch_bounds__(256) void featvol_kernel(
    const float* __restrict__ ref,
    const float* __restrict__ tgt,
    float* __restrict__ out)
{
  static_assert((W % 4) == 0, "vec4 alignment");
  __shared__ float s_ref[ROWS * W];
  __shared__ float s_tgt[ROWS * W];

  const int tid  = threadIdx.x;
  const int row0 = blockIdx.x * ROWS;            // first (b*C+c)*H + h row

  // --- Stage ROWS rows of ref and tgt into LDS: one async b128 per lane ----
  constexpr int nvecRow   = W / 4;               // 16B chunks per row
  constexpr int halfVecs  = ROWS * nvecRow;      // chunks per matrix
  constexpr int stageVecs = 2 * halfVecs;        // ref + tgt
  static_assert(stageVecs <= 256, "one async op per lane");

  if (tid < stageVecs) {
    const bool isTgt = tid >= halfVecs;
    const int  i     = isTgt ? (tid - halfVecs) : tid;   // row-major chunk id
    const float* gbase = isTgt ? tgt : ref;
    const float* g = gbase + (size_t)(row0 + i / nvecRow) * W + (i % nvecRow) * 4;
    float* l = (isTgt ? s_tgt : s_ref) + i * 4;
    async_copy16(g, l);
  }
  wait_async0();
  __syncthreads();

  // --- D x W outputs per row, vec4 NT stores -------------------------------
  constexpr int HW      = H * W;
  constexpr int rowVecs = (D * W) / 4;
  constexpr int total   = ROWS * rowVecs;

  for (int v = tid; v < total; v += 256) {
    const int r  = v / rowVecs;                  // local row in block
    const int u  = v - r * rowVecs;
    const int j  = u * 4;
    const int d  = j / W;                        // all 4 lanes share d (W%4==0)
    const int x0 = j - d * W;

    const float* sr = s_ref + r * W;
    const float* st = s_tgt + r * W;
    const v4f rv = *(const v4f*)(sr + x0);       // ds_load_b128 (16B aligned)

    v4f o;
#pragma unroll
    for (int k = 0; k < 4; ++k) {
      const int x = x0 + k;
      o[k] = (x >= d) ? fabsf(rv[k] - st[x - d]) : 0.0f;
    }

    const int row = row0 + r;                    // (b*C+c)*H + h
    const int bc  = row / H;
    const int h   = row - bc * H;
    float* dst = out + ((size_t)bc * D + d) * HW + (size_t)h * W + x0;
    __builtin_nontemporal_store(o, (v4f*)dst);   // global_store_b128 th:NT
  }
}

extern "C" void kernel_launch(void* const* d_in, const int* in_sizes, int n_in,
                              void* d_out, int out_size, void* d_ws, size_t ws_size,
                              hipStream_t stream)
{
  (void)in_sizes; (void)n_in; (void)d_ws; (void)ws_size; (void)out_size;
  float* out = (float*)d_out;

  // Inputs (setup order): conv40, conv41, conv30, conv31, conv20, conv21,
  //                       conv10, conv11, maxdisp(=384, fixed by setup).
  // Output concat order:  feat6, feat5, feat4, feat3.
  const float* conv40 = (const float*)d_in[0];
  const float* conv41 = (const float*)d_in[1];
  const float* conv30 = (const float*)d_in[2];
  const float* conv31 = (const float*)d_in[3];
  const float* conv20 = (const float*)d_in[4];
  const float* conv21 = (const float*)d_in[5];
  const float* conv10 = (const float*)d_in[6];
  const float* conv11 = (const float*)d_in[7];

  constexpr int B = 4, C = 16;
  constexpr size_t off6 = 0;
  constexpr size_t off5 = off6 + (size_t)B * C * 6  * 12 * 20;   //    92,160
  constexpr size_t off4 = off5 + (size_t)B * C * 12 * 24 * 40;   //   829,440
  constexpr size_t off3 = off4 + (size_t)B * C * 24 * 48 * 80;   // 6,727,680

  // rows = B*C*H; ROWS chosen so rows % ROWS == 0 and stageVecs <= 256.
  featvol_kernel<12, 20, 6, 8>
      <<<dim3(B * C * 12 / 8), dim3(256), 0, stream>>>(conv40, conv41, out + off6);
  featvol_kernel<24, 40, 12, 4>
      <<<dim3(B * C * 24 / 4), dim3(256), 0, stream>>>(conv30, conv31, out + off5);
  featvol_kernel<48, 80, 24, 2>
      <<<dim3(B * C * 48 / 2), dim3(256), 0, stream>>>(conv20, conv21, out + off4);
  featvol_kernel<96, 160, 48, 2>
      <<<dim3(B * C * 96 / 2), dim3(256), 0, stream>>>(conv10, conv11, out + off3);
}